// ClusteringModel_50517405335846
// MI455X (gfx1250) — compile-verified
//
#include <hip/hip_runtime.h>

// ============================================================================
// ClusteringModel on MI455X (gfx1250), full-fp32 WMMA pipeline.
//
// Shapes: I=32, J=16 -> B=512 blocks; N=256, IN=1024, D=512, H=3, C=100.
// Pipeline (all f32, V_WMMA_F32_16X16X4_F32 for every GEMM stage):
//   K1: F = x @ W_bb + b_bb                      (512 x [256x1024]@[1024x512])
//   K2: G = F^T F + eps*I                        (512 x [512x512], K=256)
//   K3: L = chol(G) (blocked, NB=64, panel in LDS); M = (L^{-1})^T row-major
//   K4: FO = F @ M * sqrt(N) (LDS), out[h] = FO @ W_heads[h] + b_heads[h]
//
// Workspace layout (floats), requires ws_size >= 1.25 GiB:
//   F : 512*256*512   = 67,108,864
//   G : 512*512*512   = 134,217,728   (Cholesky done in place -> L lower)
//   M : 512*512*512   = 134,217,728   (L^{-1} transposed, upper triangular)
// ============================================================================

typedef __attribute__((ext_vector_type(2))) float v2f;
typedef __attribute__((ext_vector_type(8))) float v8f;

#define NBLK 512
#define NN   256
#define KIN  1024
#define DD   512
#define NH   3
#define NC   100
#define EPSR 1e-7f

#define WMMA_F32(a, b, c) \
  __builtin_amdgcn_wmma_f32_16x16x4_f32(false, (a), false, (b), (short)0, (c), false, false)

// A fragment (16x4 f32): lanes 0-15 -> rows m0+r, K = k0+{0,1};
//                        lanes 16-31 -> rows m0+r, K = k0+{2,3}.
__device__ __forceinline__ v2f frag_a(const float* base, int stride, int m0, int k0,
                                      int r, int half) {
  const int kk = k0 + 2 * half;
  v2f a;
  a.x = base[(m0 + r) * stride + kk];
  a.y = base[(m0 + r) * stride + kk + 1];
  return a;
}

// B fragment (4x16 f32): vgpr0 row K = k0+2*half, vgpr1 row K = k0+2*half+1.
__device__ __forceinline__ v2f frag_b(const float* base, int stride, int k0, int n0,
                                      int r, int half) {
  const int kk = k0 + 2 * half;
  v2f b;
  b.x = base[kk * stride + n0 + r];
  b.y = base[(kk + 1) * stride + n0 + r];
  return b;
}

// ============================================================================
// K1: backbone GEMM, per-WG tile 128(M) x 128(N), K-tile 32, 8 waves.
// ============================================================================
__global__ __launch_bounds__(256) void k_backbone(const float* __restrict__ x,
                                                  const float* __restrict__ Wbb,
                                                  const float* __restrict__ bbb,
                                                  float* __restrict__ F) {
  __shared__ float As[128 * 33];
  __shared__ float Bs[32 * 132];
  const int b  = blockIdx.z;
  const int mb = blockIdx.y * 128;
  const int nb = blockIdx.x * 128;
  const int t = threadIdx.x;
  const int lane = t & 31, wid = t >> 5;
  const int r = lane & 15, half = lane >> 4;
  const int wm = (wid >> 2) * 64;  // wave row  (0,64)
  const int wn = (wid & 3) * 32;   // wave col  (0..96)
  v8f acc[4][2] = {};
  const float* xb = x + ((size_t)b * NN + mb) * KIN;

  for (int k0 = 0; k0 < KIN; k0 += 32) {
    __syncthreads();
#pragma unroll
    for (int s = 0; s < 4; ++s) {  // As[128][32] <- x tile (coalesced float4)
      int id = t + 256 * s;
      int rr = id >> 3, c4 = (id & 7) * 4;
      float4 v = *(const float4*)(xb + (size_t)rr * KIN + k0 + c4);
      As[rr * 33 + c4 + 0] = v.x; As[rr * 33 + c4 + 1] = v.y;
      As[rr * 33 + c4 + 2] = v.z; As[rr * 33 + c4 + 3] = v.w;
    }
#pragma unroll
    for (int s = 0; s < 4; ++s) {  // Bs[32][128] <- W_bb tile
      int id = t + 256 * s;
      int rr = id >> 5, c4 = (id & 31) * 4;
      float4 v = *(const float4*)(Wbb + (size_t)(k0 + rr) * DD + nb + c4);
      Bs[rr * 132 + c4 + 0] = v.x; Bs[rr * 132 + c4 + 1] = v.y;
      Bs[rr * 132 + c4 + 2] = v.z; Bs[rr * 132 + c4 + 3] = v.w;
    }
    __syncthreads();
#pragma unroll
    for (int kk = 0; kk < 32; kk += 4) {
      v2f bf0 = frag_b(Bs, 132, kk, wn, r, half);
      v2f bf1 = frag_b(Bs, 132, kk, wn + 16, r, half);
#pragma unroll
      for (int mi = 0; mi < 4; ++mi) {
        v2f af = frag_a(As, 33, wm + mi * 16, kk, r, half);
        acc[mi][0] = WMMA_F32(af, bf0, acc[mi][0]);
        acc[mi][1] = WMMA_F32(af, bf1, acc[mi][1]);
      }
    }
  }
  // epilogue: fused bias, store F (C layout: vgpr v -> row v + 8*half, col r)
#pragma unroll
  for (int mi = 0; mi < 4; ++mi) {
#pragma unroll
    for (int ni = 0; ni < 2; ++ni) {
      const int gc = nb + wn + ni * 16 + r;
      const float bias = bbb[gc];
#pragma unroll
      for (int v = 0; v < 8; ++v) {
        const int gr = mb + wm + mi * 16 + v + 8 * half;
        F[((size_t)b * NN + gr) * DD + gc] = acc[mi][ni][v] + bias;
      }
    }
  }
}

// ============================================================================
// K2: Gram G = F^T F + eps*I. A = F^T staged transposed into LDS.
// ============================================================================
__global__ __launch_bounds__(256) void k_gram(const float* __restrict__ F,
                                              float* __restrict__ G) {
  __shared__ float As[128 * 33];
  __shared__ float Bs[32 * 132];
  const int b  = blockIdx.z;
  const int db = blockIdx.y * 128;  // G rows (d)
  const int eb = blockIdx.x * 128;  // G cols (e)
  const int t = threadIdx.x;
  const int lane = t & 31, wid = t >> 5;
  const int r = lane & 15, half = lane >> 4;
  const int wm = (wid >> 2) * 64;
  const int wn = (wid & 3) * 32;
  v8f acc[4][2] = {};
  const float* Fb = F + (size_t)b * NN * DD;

  for (int n0 = 0; n0 < NN; n0 += 32) {
    __syncthreads();
#pragma unroll
    for (int s = 0; s < 4; ++s) {  // As[d_local][n_local] = F[n][d] (transpose)
      int id = t + 256 * s;
      int nl = id >> 5, c4 = (id & 31) * 4;
      float4 v = *(const float4*)(Fb + (size_t)(n0 + nl) * DD + db + c4);
      As[(c4 + 0) * 33 + nl] = v.x; As[(c4 + 1) * 33 + nl] = v.y;
      As[(c4 + 2) * 33 + nl] = v.z; As[(c4 + 3) * 33 + nl] = v.w;
    }
#pragma unroll
    for (int s = 0; s < 4; ++s) {  // Bs[n_local][e_local] = F[n][e]
      int id = t + 256 * s;
      int nl = id >> 5, c4 = (id & 31) * 4;
      float4 v = *(const float4*)(Fb + (size_t)(n0 + nl) * DD + eb + c4);
      Bs[nl * 132 + c4 + 0] = v.x; Bs[nl * 132 + c4 + 1] = v.y;
      Bs[nl * 132 + c4 + 2] = v.z; Bs[nl * 132 + c4 + 3] = v.w;
    }
    __syncthreads();
#pragma unroll
    for (int kk = 0; kk < 32; kk += 4) {
      v2f bf0 = frag_b(Bs, 132, kk, wn, r, half);
      v2f bf1 = frag_b(Bs, 132, kk, wn + 16, r, half);
#pragma unroll
      for (int mi = 0; mi < 4; ++mi) {
        v2f af = frag_a(As, 33, wm + mi * 16, kk, r, half);
        acc[mi][0] = WMMA_F32(af, bf0, acc[mi][0]);
        acc[mi][1] = WMMA_F32(af, bf1, acc[mi][1]);
      }
    }
  }
#pragma unroll
  for (int mi = 0; mi < 4; ++mi) {
#pragma unroll
    for (int ni = 0; ni < 2; ++ni) {
      const int gc = eb + wn + ni * 16 + r;
#pragma unroll
      for (int v = 0; v < 8; ++v) {
        const int gr = db + wm + mi * 16 + v + 8 * half;
        float val = acc[mi][ni][v];
        if (gr == gc) val += EPSR;
        G[((size_t)b * DD + gr) * DD + gc] = val;
      }
    }
  }
}

// ============================================================================
// K3: per-block blocked Cholesky (NB=64, panel held fully in LDS: up to
// 448x65 + 64x65 floats = 130 KiB, fits the 320 KiB WGP LDS), then
// triangular inverse written transposed: M[j][i] = (L^{-1})[i][j].
// ============================================================================
__global__ __launch_bounds__(256) void k_chol_trtri(float* __restrict__ G,
                                                    float* __restrict__ Mt) {
  extern __shared__ float sm[];
  float* Dk  = sm;             // 64*65 diag block
  float* Pan = sm + 64 * 65;   // up to 448*65 panel
  const int b = blockIdx.x;
  const int t = threadIdx.x;
  float* A = G + (size_t)b * DD * DD;
  float* M = Mt + (size_t)b * DD * DD;

  for (int kb = 0; kb < DD / 64; ++kb) {
    const int j0 = kb * 64;
    for (int idx = t; idx < 64 * 64; idx += 256) {  // load diag block
      int rr = idx >> 6, cc = idx & 63;
      Dk[rr * 65 + cc] = A[(size_t)(j0 + rr) * DD + j0 + cc];
    }
    __syncthreads();
    for (int j = 0; j < 64; ++j) {  // factor diag block (lower) in LDS
      if (t == 0) Dk[j * 65 + j] = sqrtf(Dk[j * 65 + j]);
      __syncthreads();
      const float dj = Dk[j * 65 + j];
      for (int i = j + 1 + t; i < 64; i += 256) Dk[i * 65 + j] /= dj;
      __syncthreads();
      const int w = 63 - j;
      for (int idx = t; idx < w * w; idx += 256) {
        int il = idx / w, jl = idx % w;
        int i = j + 1 + il, jj = j + 1 + jl;
        if (jj <= i) Dk[i * 65 + jj] -= Dk[i * 65 + j] * Dk[jj * 65 + j];
      }
      __syncthreads();
    }
    for (int idx = t; idx < 64 * 64; idx += 256) {  // write diag back (lower)
      int rr = idx >> 6, cc = idx & 63;
      if (cc <= rr) A[(size_t)(j0 + rr) * DD + j0 + cc] = Dk[rr * 65 + cc];
    }
    const int ntr = DD - j0 - 64;
    if (ntr > 0) {
      for (int idx4 = t; idx4 < ntr * 16; idx4 += 256) {  // load panel to LDS
        int rr = idx4 >> 4, c4 = (idx4 & 15) * 4;
        float4 v = *(const float4*)(A + (size_t)(j0 + 64 + rr) * DD + j0 + c4);
        Pan[rr * 65 + c4 + 0] = v.x; Pan[rr * 65 + c4 + 1] = v.y;
        Pan[rr * 65 + c4 + 2] = v.z; Pan[rr * 65 + c4 + 3] = v.w;
      }
      __syncthreads();
      // panel TRSM: row p solves p * Lkk^T = g (forward substitution)
      for (int pr = t; pr < ntr; pr += 256) {
        float* P = Pan + pr * 65;
        for (int c = 0; c < 64; ++c) {
          float v = P[c];
          for (int tt = 0; tt < c; ++tt) v -= P[tt] * Dk[c * 65 + tt];
          P[c] = v / Dk[c * 65 + c];
        }
      }
      __syncthreads();
      for (int idx4 = t; idx4 < ntr * 16; idx4 += 256) {  // write panel back
        int rr = idx4 >> 4, c4 = (idx4 & 15) * 4;
        float4 v;
        v.x = Pan[rr * 65 + c4 + 0]; v.y = Pan[rr * 65 + c4 + 1];
        v.z = Pan[rr * 65 + c4 + 2]; v.w = Pan[rr * 65 + c4 + 3];
        *(float4*)(A + (size_t)(j0 + 64 + rr) * DD + j0 + c4) = v;
      }
      // trailing syrk update (lower) from the LDS-resident panel
      for (int idx = t; idx < ntr * ntr; idx += 256) {
        int il = idx / ntr, jl = idx % ntr;
        if (jl > il) continue;
        float s = 0.f;
        for (int c = 0; c < 64; ++c) s += Pan[il * 65 + c] * Pan[jl * 65 + c];
        A[(size_t)(j0 + 64 + il) * DD + (j0 + 64 + jl)] -= s;
      }
    }
    __syncthreads();
  }
  // trtri: column j of L^{-1} -> row j of M (so K4 reads M row-major).
  for (int j = t; j < DD; j += 256) {
    float* Mr = M + (size_t)j * DD;
    for (int i = 0; i < j; ++i) Mr[i] = 0.f;
    Mr[j] = 1.0f / A[(size_t)j * DD + j];
    for (int i = j + 1; i < DD; ++i) {
      float s = 0.f;
      const float* Ai = A + (size_t)i * DD;
      for (int tt = j; tt < i; ++tt) s += Ai[tt] * Mr[tt];
      Mr[i] = -s / Ai[i];
    }
  }
}

// ============================================================================
// K4: FO = (F tile) @ M * sqrt(N) into LDS (skipping zero tiles of the upper-
// triangular M), then out[h] = FO @ W_heads[h] + b_heads[h]. LDS: 2x64x516 f32.
// ============================================================================
__global__ __launch_bounds__(256) void k_ortho_heads(const float* __restrict__ F,
                                                     const float* __restrict__ Mt,
                                                     const float* __restrict__ Wh,
                                                     const float* __restrict__ bh,
                                                     float* __restrict__ out) {
  extern __shared__ float sm[];
  float* Ftl = sm;             // 64*516: f tile
  float* FO  = sm + 64 * 516;  // 64*516: orthonormalized tile
  const int b  = blockIdx.y;
  const int nb = blockIdx.x * 64;
  const int t = threadIdx.x;
  const int lane = t & 31, wid = t >> 5;
  const int r = lane & 15, half = lane >> 4;

  const float* Fb = F + ((size_t)b * NN + nb) * DD;
#pragma unroll
  for (int s = 0; s < 32; ++s) {  // stage f tile (64x512)
    int id = t + 256 * s;
    int rr = id >> 7, c4 = (id & 127) * 4;
    float4 v = *(const float4*)(Fb + (size_t)rr * DD + c4);
    Ftl[rr * 516 + c4 + 0] = v.x; Ftl[rr * 516 + c4 + 1] = v.y;
    Ftl[rr * 516 + c4 + 2] = v.z; Ftl[rr * 516 + c4 + 3] = v.w;
  }
  __syncthreads();

  const float* Mg = Mt + (size_t)b * DD * DD;
#pragma unroll
  for (int et = 0; et < 4; ++et) {  // phase 1: FO = Ftl @ M * 16
    const int e0 = wid * 64 + et * 16;
    v8f acc[4] = {};
    for (int k0 = 0; k0 < e0 + 16; k0 += 4) {  // M upper-tri: skip k > e
      v2f bf = frag_b(Mg, DD, k0, e0, r, half);
#pragma unroll
      for (int mi = 0; mi < 4; ++mi) {
        v2f af = frag_a(Ftl, 516, mi * 16, k0, r, half);
        acc[mi] = WMMA_F32(af, bf, acc[mi]);
      }
    }
#pragma unroll
    for (int mi = 0; mi < 4; ++mi)
#pragma unroll
      for (int v = 0; v < 8; ++v)
        FO[(mi * 16 + v + 8 * half) * 516 + e0 + r] = acc[mi][v] * 16.0f;  // sqrt(256)
  }
  __syncthreads();

  // phase 2: head GEMMs out of LDS. 84 C-tiles = 3 heads x 4 m x 7 c.
  for (int tile = wid; tile < 84; tile += 8) {
    const int h = tile / 28, rem = tile % 28;
    const int ct = rem >> 2, mi = rem & 3;
    const int cc = ct * 16 + r;
    const bool cv = (cc < NC);
    const float* W = Wh + (size_t)h * DD * NC;
    v8f acc = {};
    for (int k0 = 0; k0 < DD; k0 += 4) {
      const int kk = k0 + 2 * half;
      v2f bf;
      bf.x = cv ? W[kk * NC + cc] : 0.f;
      bf.y = cv ? W[(kk + 1) * NC + cc] : 0.f;
      v2f af = frag_a(FO, 516, mi * 16, k0, r, half);
      acc = WMMA_F32(af, bf, acc);
    }
    if (cv) {
      const float bias = bh[h * NC + cc];
#pragma unroll
      for (int v = 0; v < 8; ++v) {
        const int n = nb + mi * 16 + v + 8 * half;
        out[(((size_t)h * NBLK + b) * NN + n) * NC + cc] = acc[v] + bias;
      }
    }
  }
}

// ============================================================================
extern "C" void kernel_launch(void* const* d_in, const int* in_sizes, int n_in,
                              void* d_out, int out_size, void* d_ws, size_t ws_size,
                              hipStream_t stream) {
  const float* x   = (const float*)d_in[0];
  const float* Wbb = (const float*)d_in[1];
  const float* bbb = (const float*)d_in[2];
  const float* Wh  = (const float*)d_in[3];
  const float* bh  = (const float*)d_in[4];
  float* out = (float*)d_out;

  // workspace: F (256 MiB) | G (512 MiB) | M (512 MiB)
  float* F = (float*)d_ws;
  float* G = F + (size_t)NBLK * NN * DD;
  float* M = G + (size_t)NBLK * DD * DD;

  k_backbone<<<dim3(DD / 128, NN / 128, NBLK), 256, 0, stream>>>(x, Wbb, bbb, F);
  k_gram<<<dim3(DD / 128, DD / 128, NBLK), 256, 0, stream>>>(F, G);
  k_chol_trtri<<<dim3(NBLK), 256, (64 * 65 + 448 * 65) * sizeof(float), stream>>>(G, M);
  k_ortho_heads<<<dim3(NN / 64, NBLK), 256, (size_t)2 * 64 * 516 * sizeof(float), stream>>>(
      F, M, Wh, bh, out);
}